// NaiveDFTQNN_17738214932736
// MI455X (gfx1250) — compile-verified
//
#include <hip/hip_runtime.h>
#include <math.h>
#include <stdint.h>

#define DIM (1 << 25)
#define NPART 512

typedef __attribute__((ext_vector_type(2))) float v2f;
typedef __attribute__((ext_vector_type(8))) float v8f;

// ws layout (floats):
//   ws[0]              = cos(theta0/2) / ||feature||
//   ws[1+2g], ws[2+2g] = cos, sin of theta[g]/2 ; gate g acts on bits (23-2g, 22-2g)
// Norm partial sums are parked in d_out[0..511] (overwritten later by pass_low).

// ---- CDNA5 async global<->LDS helpers (ASYNCcnt-tracked) --------------------
__device__ __forceinline__ uint32_t lds_offset_of(const void* p) {
  // Generic LDS pointers carry the aperture in the high 32 bits; low 32 bits
  // are the raw LDS byte address (ISA 10.2 aperture decomposition).
  return (uint32_t)(uintptr_t)p;
}

__device__ __forceinline__ void async_load_lds_b128(uint32_t lds_addr,
                                                    uint32_t voff,
                                                    const void* sbase) {
  asm volatile("global_load_async_to_lds_b128 %0, %1, %2"
               :
               : "v"(lds_addr), "v"(voff), "s"(sbase)
               : "memory");
}

__device__ __forceinline__ void async_store_lds_b128(uint32_t voff,
                                                     uint32_t lds_addr,
                                                     void* sbase) {
  asm volatile("global_store_async_from_lds_b128 %0, %1, %2"
               :
               : "v"(voff), "v"(lds_addr), "s"(sbase)
               : "memory");
}

__device__ __forceinline__ void wait_asynccnt0() {
  asm volatile("s_wait_asynccnt 0x0" ::: "memory");
}
// -----------------------------------------------------------------------------

__global__ void __launch_bounds__(256)
norm_partial_kernel(const float* __restrict__ f, float* __restrict__ part) {
  __shared__ float red[256];
  int tid = threadIdx.x;
  int gid = blockIdx.x * 256 + tid;
  float acc = 0.f;
  for (int i = gid; i < DIM; i += NPART * 256) {
    float v = f[i];
    acc += v * v;
  }
  red[tid] = acc;
  __syncthreads();
  for (int off = 128; off > 0; off >>= 1) {
    if (tid < off) red[tid] += red[tid + off];
    __syncthreads();
  }
  if (tid == 0) part[blockIdx.x] = red[0];
}

__global__ void __launch_bounds__(512)
norm_final_kernel(const float* __restrict__ part, const float* __restrict__ theta,
                  float* __restrict__ ws) {
  __shared__ float red[NPART];
  int tid = threadIdx.x;
  red[tid] = part[tid];
  __syncthreads();
  for (int off = 256; off > 0; off >>= 1) {
    if (tid < off) red[tid] += red[tid + off];
    __syncthreads();
  }
  if (tid == 0) {
    float nrm = sqrtf(red[0]);
    ws[0] = cosf(0.5f * theta[0]) / nrm;
    for (int g = 0; g < 12; ++g) {
      float th = theta[g];
      ws[1 + 2 * g] = cosf(0.5f * th);
      ws[2 + 2 * g] = sinf(0.5f * th);
    }
  }
}

// 4x4 single-excitation matrix element: basis (b_hi b_lo) = 00,01,10,11
__device__ __forceinline__ float pair_elem(float c, float s, int a, int b) {
  if (a == 0 && b == 0) return 1.f;
  if (a == 3 && b == 3) return 1.f;
  if (a == 1 && b == 1) return c;
  if (a == 1 && b == 2) return -s;
  if (a == 2 && b == 1) return s;
  if (a == 2 && b == 2) return c;
  return 0.f;
}

// Pass low: gates on bits 6..11 (LDS butterflies) + bits 0..5 (WMMA GEMM with
// scaled 64x64 Kronecker matrix). src = feature, dst = out.
__global__ void __launch_bounds__(256)
pass_low_kernel(const float* __restrict__ src, float* __restrict__ dst,
                const float* __restrict__ ws) {
  __shared__ float tile[8192];     // 128 rows x 64 cols (bits 0..12 of index)
  __shared__ float m64[64 * 64];   // scaled M64[j][k]
  int tid = threadIdx.x;
  int base = blockIdx.x * 8192;

  // Kick off async global->LDS copy of the contiguous 32KB tile.
  {
    const float* gsrc = src + base;
    uint32_t tlds = lds_offset_of(&tile[0]);
#pragma unroll
    for (int i = 0; i < 8; ++i) {
      uint32_t lin4 = (uint32_t)(tid + i * 256);
      async_load_lds_b128(tlds + lin4 * 16u, lin4 * 16u, gsrc);
    }
  }

  // Overlap the copy: build scaled M64 for gates g=9 (bits 5,4), g=10 (bits
  // 3,2), g=11 (bits 1,0) while the TDM-path transfer is in flight.
  float scale = ws[0];
  float c9 = ws[1 + 18], s9 = ws[2 + 18];
  float c10 = ws[1 + 20], s10 = ws[2 + 20];
  float c11 = ws[1 + 22], s11 = ws[2 + 22];
  for (int e = tid; e < 4096; e += 256) {
    int j = e >> 6, k = e & 63;
    float p0 = pair_elem(c9, s9, (j >> 4) & 3, (k >> 4) & 3);
    float p1 = pair_elem(c10, s10, (j >> 2) & 3, (k >> 2) & 3);
    float p2 = pair_elem(c11, s11, j & 3, k & 3);
    m64[e] = scale * p0 * p1 * p2;
  }
  wait_asynccnt0();
  __syncthreads();

  // LDS butterflies for bits (7,6)->g8, (9,8)->g7, (11,10)->g6
  {
    const int bs[3] = {6, 8, 10};
    const int gs[3] = {8, 7, 6};
    for (int gi = 0; gi < 3; ++gi) {
      int b = bs[gi];
      float cc = ws[1 + 2 * gs[gi]], ss = ws[2 + 2 * gs[gi]];
      for (int t = tid; t < 2048; t += 256) {
        int low = t & ((1 << b) - 1);
        int up = t >> b;
        int i01 = (up << (b + 2)) | (1 << b) | low;
        int i10 = i01 + (1 << b);
        float a01 = tile[i01], a10 = tile[i10];
        tile[i01] = cc * a01 - ss * a10;
        tile[i10] = ss * a01 + cc * a10;
      }
      __syncthreads();
    }
  }

  // WMMA GEMM: out[r][j] = sum_k tile[r][k] * m64[j][k]
  // A (16x4 f32): lane l<16 -> {K=k0, K=k0+1}, lane>=16 -> {K=k0+2, K=k0+3}, M = l%16
  // B (4x16 f32): VGPR0 = K rows {k0,k0+2}, VGPR1 = {k0+1,k0+3}; N = l%16
  // D (16x16 f32): VGPR j -> M = j + 8*(l/16), N = l%16
  int lane = tid & 31;
  int wave = tid >> 5;
  int hh = lane >> 4;
  int l16 = lane & 15;
  int rowbase = wave * 16;  // 8 waves cover the 128 rows
  for (int ct = 0; ct < 4; ++ct) {
    int c0 = ct * 16;
    v8f acc = {};
#pragma unroll
    for (int k0 = 0; k0 < 64; k0 += 4) {
      v2f a = *(const v2f*)&tile[(rowbase + l16) * 64 + k0 + 2 * hh];
      v2f b = *(const v2f*)&m64[(c0 + l16) * 64 + k0 + 2 * hh];
      acc = __builtin_amdgcn_wmma_f32_16x16x4_f32(
          false, a, false, b, (short)0, acc, false, false);
    }
#pragma unroll
    for (int j = 0; j < 8; ++j) {
      dst[base + (rowbase + j + 8 * hh) * 64 + c0 + l16] = acc[j];
    }
  }
}

// Pass hi: gates on 6 high bit positions via strided 64x64 LDS tiles.
// LOG2STRIDE=12 -> h spans bits 12..17, gates g=3,4,5 (GBASE=3)
// LOG2STRIDE=18 -> h spans bits 18..23, gates g=0,1,2 (GBASE=0)
template <int LOG2STRIDE, int GBASE>
__global__ void __launch_bounds__(256)
pass_hi_kernel(float* __restrict__ data, const float* __restrict__ ws) {
  __shared__ float tile[64 * 64];
  int tid = threadIdx.x;
  int o = blockIdx.x;  // 8192 blocks
  int outer;
  if (LOG2STRIDE == 12) {
    outer = ((o >> 6) << 18) | ((o & 63) << 6);      // fixed bits 18..24 | bits 6..11
  } else {
    outer = ((o >> 12) << 24) | ((o & 4095) << 6);   // fixed bit 24 | bits 6..17
  }
  const int strideH = 1 << LOG2STRIDE;
  float* gbase = data + outer;
  uint32_t tlds = lds_offset_of(&tile[0]);

  // Async global->LDS tile gather: 64 rows of 256B at stride 4*strideH bytes.
#pragma unroll
  for (int i = 0; i < 4; ++i) {
    int lin4 = tid + i * 256;  // [0,1024): h = lin4/16, 16 float4 per 256B row
    int h = lin4 >> 4;
    int c4 = lin4 & 15;
    uint32_t voff = (uint32_t)((h * strideH + c4 * 4) * 4);
    async_load_lds_b128(tlds + (uint32_t)lin4 * 16u, voff, gbase);
  }
  wait_asynccnt0();
  __syncthreads();

#pragma unroll
  for (int gi = 0; gi < 3; ++gi) {
    int b = 4 - 2 * gi;  // h-bit pair (b+1, b)
    float cc = ws[1 + 2 * (GBASE + gi)], ss = ws[2 + 2 * (GBASE + gi)];
    for (int t = tid; t < 1024; t += 256) {
      int p = t >> 6;  // pair id [0,16)
      int c = t & 63;
      int low = p & ((1 << b) - 1);
      int up = p >> b;
      int h01 = (up << (b + 2)) | (1 << b) | low;
      int h10 = h01 + (1 << b);
      float a01 = tile[h01 * 64 + c], a10 = tile[h10 * 64 + c];
      tile[h01 * 64 + c] = cc * a01 - ss * a10;
      tile[h10 * 64 + c] = ss * a01 + cc * a10;
    }
    __syncthreads();
  }

  // Async LDS->global scatter-back; drains at s_endpgm's implicit wait-idle.
#pragma unroll
  for (int i = 0; i < 4; ++i) {
    int lin4 = tid + i * 256;
    int h = lin4 >> 4;
    int c4 = lin4 & 15;
    uint32_t voff = (uint32_t)((h * strideH + c4 * 4) * 4);
    async_store_lds_b128(voff, tlds + (uint32_t)lin4 * 16u, gbase);
  }
}

extern "C" void kernel_launch(void* const* d_in, const int* in_sizes, int n_in,
                              void* d_out, int out_size, void* d_ws, size_t ws_size,
                              hipStream_t stream) {
  const float* feature = (const float*)d_in[0];
  const float* theta = (const float*)d_in[1];
  float* out = (float*)d_out;
  float* ws = (float*)d_ws;

  // 1) deterministic two-stage norm reduction; partials parked in d_out
  norm_partial_kernel<<<NPART, 256, 0, stream>>>(feature, out);
  norm_final_kernel<<<1, NPART, 0, stream>>>(out, theta, ws);
  // 2) gates on bits 0..11 (+ global scale) via async-LDS load, LDS
  //    butterflies and f32 WMMA GEMM
  pass_low_kernel<<<DIM / 8192, 256, 0, stream>>>(feature, out, ws);
  // 3) gates on bits 12..17, then 18..23, via strided async-LDS tiles (in place)
  pass_hi_kernel<12, 3><<<8192, 256, 0, stream>>>(out, ws);
  pass_hi_kernel<18, 0><<<8192, 256, 0, stream>>>(out, ws);
}